// HypergraphModel_67405216743648
// MI455X (gfx1250) — compile-verified
//
#include <hip/hip_runtime.h>

// Problem constants (match reference setup_inputs()).
#define NNODES 100000
#define NEDGES 20000
#define NPINS  1600000
#define DIN    128
#define DHID   128
#define DOUTF  64
#define EPSLN  1e-5f
#define NEG_SLOPE 0.01f

typedef float v2f __attribute__((ext_vector_type(2)));
typedef float v8f __attribute__((ext_vector_type(8)));

// ---------------------------------------------------------------------------
// Utility: zero a float buffer.
// ---------------------------------------------------------------------------
__global__ void k_zero(float* __restrict__ p, int n) {
  int i = blockIdx.x * blockDim.x + threadIdx.x;
  if (i < n) p[i] = 0.0f;
}

// ---------------------------------------------------------------------------
// Degree accumulation: one thread per pin, native f32 atomic adds.
// Counts are integers < 2^24 so f32 accumulation is exact.
// ---------------------------------------------------------------------------
__global__ void k_degrees(const int* __restrict__ node_idx,
                          const int* __restrict__ edge_idx,
                          float* __restrict__ ddeg,   // [NNODES]
                          float* __restrict__ bdeg) { // [NEDGES]
  int i = blockIdx.x * blockDim.x + threadIdx.x;
  if (i >= NPINS) return;
  unsafeAtomicAdd(&ddeg[node_idx[i]], 1.0f);
  unsafeAtomicAdd(&bdeg[edge_idx[i]], 1.0f);
}

__global__ void k_invert(float* __restrict__ p, int n) {
  int i = blockIdx.x * blockDim.x + threadIdx.x;
  if (i >= n) return;
  float v = p[i];
  p[i] = (v > 0.0f) ? (1.0f / v) : 0.0f;
}

// ---------------------------------------------------------------------------
// f32 WMMA GEMM: C[M,Ncol] = A[M,K] @ B[K,Ncol]   (row-major, no bias)
// One wave computes one 16x16 tile via V_WMMA_F32_16X16X4_F32, stepping K by 4.
//
// A fragment (16x4, MxK) per ISA layout: lanes 0-15 hold M=lane, K=k0..k0+1;
// lanes 16-31 hold M=lane-16, K=k0+2..k0+3. B (4x16, KxN) mirrored.
// C/D: 8 VGPRs; lanes 0-15 -> rows 0..7, lanes 16-31 -> rows 8..15, col=lane%16.
// ---------------------------------------------------------------------------
__global__ void k_gemm_wmma_f32(const float* __restrict__ A,
                                const float* __restrict__ B,
                                float* __restrict__ C,
                                int M, int K, int Ncol) {
  const int gwave  = (blockIdx.x * blockDim.x + threadIdx.x) >> 5;
  const int lane   = threadIdx.x & 31;
  const int tilesN = Ncol >> 4;
  const int tm     = gwave / tilesN;
  const int tn     = gwave - tm * tilesN;
  if (tm * 16 >= M) return;

  const int mrow = tm * 16 + (lane & 15);  // A row for this lane
  const int ncol = tn * 16 + (lane & 15);  // B/C column for this lane
  const int kh   = (lane >> 4) << 1;       // 0 for lanes 0-15, 2 for lanes 16-31

  v8f acc = {0.f, 0.f, 0.f, 0.f, 0.f, 0.f, 0.f, 0.f};

  const float* arow = A + (size_t)mrow * K;
  for (int k0 = 0; k0 < K; k0 += 4) {
    // A: two contiguous f32 (8B aligned: K=128, k0%4==0, kh even)
    v2f a = *(const v2f*)(arow + k0 + kh);
    v2f b;
    b.x = B[(size_t)(k0 + kh)     * Ncol + ncol];
    b.y = B[(size_t)(k0 + kh + 1) * Ncol + ncol];
    acc = __builtin_amdgcn_wmma_f32_16x16x4_f32(
        /*neg_a=*/false, a, /*neg_b=*/false, b,
        /*c_mod=*/(short)0, acc, /*reuse_a=*/false, /*reuse_b=*/false);
  }

  const int row0 = tm * 16 + ((lane >> 4) << 3);
#pragma unroll
  for (int v = 0; v < 8; ++v)
    C[(size_t)(row0 + v) * Ncol + ncol] = acc[v];
}

// ---------------------------------------------------------------------------
// Segment scatter-add: one wave per pin, FPL floats per lane (F = 32*FPL).
//   dst[dst_idx[p]][:] += src[src_idx[p]][:] * (scale ? scale[src_idx[p]] : 1)
// Folding `scale` (Binv) into the gather removes a full elementwise pass.
// All rows are L2-resident (x/h 51MB, edge accum 10MB << 192MB L2).
// ---------------------------------------------------------------------------
template <int FPL>
__global__ void k_scatter(const float* __restrict__ src,
                          const int* __restrict__ src_idx,
                          const int* __restrict__ dst_idx,
                          const float* __restrict__ scale,  // nullable
                          float* __restrict__ dst) {
  const int pin  = (blockIdx.x * blockDim.x + threadIdx.x) >> 5;
  const int lane = threadIdx.x & 31;
  if (pin >= NPINS) return;
  const int s = src_idx[pin];
  const int d = dst_idx[pin];
  const float sc = scale ? scale[s] : 1.0f;
  const float* sp = src + (size_t)s * (FPL * 32) + lane * FPL;
  float*       dp = dst + (size_t)d * (FPL * 32) + lane * FPL;
#pragma unroll
  for (int j = 0; j < FPL; ++j)
    unsafeAtomicAdd(&dp[j], sp[j] * sc);
}

// ---------------------------------------------------------------------------
// Fused node finalize for conv1: h = h*Dinv + b1, LayerNorm, LeakyReLU.
// One wave per node row (128 feats, 4/lane), wave32 shuffle reductions.
// ---------------------------------------------------------------------------
__global__ void k_node_ln(float* __restrict__ h,
                          const float* __restrict__ dinv,
                          const float* __restrict__ b1,
                          const float* __restrict__ gamma,
                          const float* __restrict__ beta) {
  const int row  = (blockIdx.x * blockDim.x + threadIdx.x) >> 5;
  const int lane = threadIdx.x & 31;
  if (row >= NNODES) return;
  float* rp = h + (size_t)row * DHID;
  const float di = dinv[row];

  float v[4];
  float s = 0.0f;
#pragma unroll
  for (int j = 0; j < 4; ++j) {
    const int c = lane * 4 + j;
    v[j] = rp[c] * di + b1[c];
    s += v[j];
  }
#pragma unroll
  for (int off = 16; off > 0; off >>= 1) s += __shfl_xor(s, off, 32);
  const float mu = s * (1.0f / DHID);

  float var = 0.0f;
#pragma unroll
  for (int j = 0; j < 4; ++j) {
    const float d = v[j] - mu;
    var += d * d;
  }
#pragma unroll
  for (int off = 16; off > 0; off >>= 1) var += __shfl_xor(var, off, 32);
  const float inv = rsqrtf(var * (1.0f / DHID) + EPSLN);

#pragma unroll
  for (int j = 0; j < 4; ++j) {
    const int c = lane * 4 + j;
    float t = (v[j] - mu) * inv * gamma[c] + beta[c];
    rp[c] = (t > 0.0f) ? t : NEG_SLOPE * t;
  }
}

// ---------------------------------------------------------------------------
// Output finalize: out = out*Dinv[row] + b3[col]   (N x 64)
// ---------------------------------------------------------------------------
__global__ void k_finalize_out(float* __restrict__ out,
                               const float* __restrict__ dinv,
                               const float* __restrict__ b3) {
  int i = blockIdx.x * blockDim.x + threadIdx.x;
  if (i >= NNODES * DOUTF) return;
  const int r = i / DOUTF;
  const int c = i - r * DOUTF;
  out[i] = out[i] * dinv[r] + b3[c];
}

// ---------------------------------------------------------------------------
extern "C" void kernel_launch(void* const* d_in, const int* in_sizes, int n_in,
                              void* d_out, int out_size, void* d_ws, size_t ws_size,
                              hipStream_t stream) {
  const float* x     = (const float*)d_in[0];
  const int*   hei   = (const int*)d_in[1];   // [2, NNZ]; integer -> const int*
  const int* node_idx = hei;
  const int* edge_idx = hei + NPINS;
  const float* W1    = (const float*)d_in[2];
  const float* b1    = (const float*)d_in[3];
  const float* gamma = (const float*)d_in[4];
  const float* beta  = (const float*)d_in[5];
  const float* W3    = (const float*)d_in[6];
  const float* b3    = (const float*)d_in[7];
  float* out = (float*)d_out;

  // Workspace layout (floats): ~113 MB total.
  float* buf_y = (float*)d_ws;                          // N*128 (GEMM out, reused N*64)
  float* buf_h = buf_y + (size_t)NNODES * DHID;         // N*128 (node accum / h)
  float* buf_e = buf_h + (size_t)NNODES * DHID;         // E*128 (edge accum, reused E*64)
  float* dinv  = buf_e + (size_t)NEDGES * DHID;         // N
  float* binv  = dinv + NNODES;                         // E (contiguous after dinv)

  const int TB = 256;
  const int pinWaveBlocks  = (NPINS * 32 + TB - 1) / TB;   // one wave per pin
  const int nodeWaveBlocks = (NNODES * 32 + TB - 1) / TB;  // one wave per node

  // --- degrees -> Dinv / Binv ---
  k_zero<<<(NNODES + NEDGES + TB - 1) / TB, TB, 0, stream>>>(dinv, NNODES + NEDGES);
  k_degrees<<<(NPINS + TB - 1) / TB, TB, 0, stream>>>(node_idx, edge_idx, dinv, binv);
  k_invert<<<(NNODES + NEDGES + TB - 1) / TB, TB, 0, stream>>>(dinv, NNODES + NEDGES);

  // --- conv1: y = x @ W1 (f32 WMMA) ---
  {
    const int waves = (NNODES / 16) * (DHID / 16);  // 50000
    k_gemm_wmma_f32<<<waves * 32 / TB, TB, 0, stream>>>(x, W1, buf_y, NNODES, DIN, DHID);
  }
  // node -> edge
  k_zero<<<(NEDGES * DHID + TB - 1) / TB, TB, 0, stream>>>(buf_e, NEDGES * DHID);
  k_scatter<4><<<pinWaveBlocks, TB, 0, stream>>>(buf_y, node_idx, edge_idx, nullptr, buf_e);
  // edge -> node, with Binv folded into the gather
  k_zero<<<(NNODES * DHID + TB - 1) / TB, TB, 0, stream>>>(buf_h, NNODES * DHID);
  k_scatter<4><<<pinWaveBlocks, TB, 0, stream>>>(buf_e, edge_idx, node_idx, binv, buf_h);
  // h = LeakyReLU(LayerNorm(h*Dinv + b1))
  k_node_ln<<<nodeWaveBlocks, TB, 0, stream>>>(buf_h, dinv, b1, gamma, beta);

  // --- conv2: y2 = h @ W3 (f32 WMMA) ---
  {
    const int waves = (NNODES / 16) * (DOUTF / 16);  // 25000
    k_gemm_wmma_f32<<<waves * 32 / TB, TB, 0, stream>>>(buf_h, W3, buf_y, NNODES, DHID, DOUTF);
  }
  // node -> edge
  k_zero<<<(NEDGES * DOUTF + TB - 1) / TB, TB, 0, stream>>>(buf_e, NEDGES * DOUTF);
  k_scatter<2><<<pinWaveBlocks, TB, 0, stream>>>(buf_y, node_idx, edge_idx, nullptr, buf_e);
  // edge -> node into d_out, with Binv folded
  k_zero<<<(NNODES * DOUTF + TB - 1) / TB, TB, 0, stream>>>(out, NNODES * DOUTF);
  k_scatter<2><<<pinWaveBlocks, TB, 0, stream>>>(buf_e, edge_idx, node_idx, binv, out);
  // out = out*Dinv + b3
  k_finalize_out<<<(NNODES * DOUTF + TB - 1) / TB, TB, 0, stream>>>(out, dinv, b3);
}